// PiecewiseLinearMLP_13108240187538
// MI455X (gfx1250) — compile-verified
//
#include <hip/hip_runtime.h>
#include <hip/hip_bf16.h>

// ---------------------------------------------------------------------------
// PiecewiseLinearMLP for MI455X (gfx1250, wave32, WMMA, async-to-LDS).
//
// contribution[b,i] = sum_{j,k} effw[i,j]*w_pw[i,j,k]*relu(x[b,j]-bp[i,j,k])
//                   + sum_j effw[i,j]*b_pw[i,j]
// out[b,i,{0,1}]   = relu(c*W1[i]+b1[i]) @ W2[i]^T + b2[i]
//
// VALU computes the piecewise activations (true bottleneck; each element is
// used once, so matrix units cannot replace that work). V_WMMA_F32_16X16X4_F32
// (f32 operands, TRANS-class -> co-executes with VALU) performs the
// j-reduction: A = 16(b) x 4(j) piecewise values, B = ones-in-column-0,
// D accumulates contributions in column 0 (two interleaved accumulators to
// break the D->C RAW chain). The per-node MLP is a genuine (16x32)*(32x2)
// WMMA GEMM. x tile staged via GLOBAL_LOAD_ASYNC_TO_LDS_B128 (ASYNCcnt).
// ---------------------------------------------------------------------------

#define NN      256   // num nodes
#define PP      5     // pieces
#define HH      32    // hidden
#define BB      256   // batch
#define BCHUNK  128   // batch rows per block
#define XSTRIDE 260   // x tile LDS stride: 16B-aligned rows, conflict-free b64
#define THRESH  0.01f

typedef float v2f __attribute__((ext_vector_type(2)));
typedef float v8f __attribute__((ext_vector_type(8)));
typedef int   v4i __attribute__((ext_vector_type(4)));
typedef __attribute__((address_space(1))) v4i* gv4i_p;  // global <4 x i32>*
typedef __attribute__((address_space(3))) v4i* lv4i_p;  // LDS    <4 x i32>*

#if __has_builtin(__builtin_amdgcn_global_load_async_to_lds_b128) && \
    __has_builtin(__builtin_amdgcn_s_wait_asynccnt)
#define USE_ASYNC_LDS 1
#else
#define USE_ASYNC_LDS 0
#endif

__device__ __forceinline__ v8f wmma4(v2f a, v2f b, v8f c) {
  // (neg_a, A, neg_b, B, c_mod, C, reuse_a, reuse_b)
  return __builtin_amdgcn_wmma_f32_16x16x4_f32(
      false, a, false, b, (short)0, c, false, false);
}

// cs record per column j (16 floats, 64B): [w0..w3 | bp0..bp3 | w4 bp4 | pad]
// Evaluates sum_k w[j,k]*relu(xv - bp[j,k]) for columns j and j+1 (j even).
__device__ __forceinline__ v2f pwc_pair(const float* __restrict__ xs_row,
                                        const float* __restrict__ cs,
                                        int j) {
  const float2 xv = *reinterpret_cast<const float2*>(xs_row + j);  // b64
  v2f r;
  {
    const float4 w = *reinterpret_cast<const float4*>(cs + j * 16);
    const float4 p = *reinterpret_cast<const float4*>(cs + j * 16 + 4);
    const float2 t = *reinterpret_cast<const float2*>(cs + j * 16 + 8);
    float s;
    s  = w.x * fmaxf(xv.x - p.x, 0.f);
    s += w.y * fmaxf(xv.x - p.y, 0.f);
    s += w.z * fmaxf(xv.x - p.z, 0.f);
    s += w.w * fmaxf(xv.x - p.w, 0.f);
    s += t.x * fmaxf(xv.x - t.y, 0.f);
    r.x = s;
  }
  {
    const float4 w = *reinterpret_cast<const float4*>(cs + (j + 1) * 16);
    const float4 p = *reinterpret_cast<const float4*>(cs + (j + 1) * 16 + 4);
    const float2 t = *reinterpret_cast<const float2*>(cs + (j + 1) * 16 + 8);
    float s;
    s  = w.x * fmaxf(xv.y - p.x, 0.f);
    s += w.y * fmaxf(xv.y - p.y, 0.f);
    s += w.z * fmaxf(xv.y - p.z, 0.f);
    s += w.w * fmaxf(xv.y - p.w, 0.f);
    s += t.x * fmaxf(xv.y - t.y, 0.f);
    r.y = s;
  }
  return r;
}

__global__ __launch_bounds__(256)
void pwl_mlp_kernel(const float* __restrict__ x,
                    const float* __restrict__ adjacency,
                    const float* __restrict__ bp,
                    const float* __restrict__ w_pw,
                    const float* __restrict__ b_pw,
                    const float* __restrict__ W1,
                    const float* __restrict__ b1,
                    const float* __restrict__ W2,
                    const float* __restrict__ b2,
                    float* __restrict__ out) {
  __shared__ float xs[BCHUNK * XSTRIDE];   // x tile, [b][j]
  __shared__ float cs[NN * 16];            // fused coeffs per column
  __shared__ float red[NN];                // bias tree-reduction
  __shared__ float carr[BCHUNK];           // contributions
  __shared__ float W1s[HH], b1s[HH], W2s[2 * HH], b2s[2];

  const int tid     = threadIdx.x;
  const int i       = blockIdx.x;          // target node
  const int bchunk0 = blockIdx.y * BCHUNK; // batch base

  // ---------------- Phase 0: stage per-i data into LDS ----------------
#if USE_ASYNC_LDS
  // Async DMA of the x tile: global -> LDS without touching VGPRs.
  for (int e = tid; e < BCHUNK * NN / 4; e += 256) {
    const int bb = e >> 6;           // NN/4 = 64 b128 chunks per row
    const int j4 = (e & 63) << 2;
    __builtin_amdgcn_global_load_async_to_lds_b128(
        (gv4i_p)(x + (size_t)(bchunk0 + bb) * NN + j4),
        (lv4i_p)(xs + bb * XSTRIDE + j4),
        0, 0);
  }
#else
  for (int e = tid; e < BCHUNK * NN; e += 256) {
    const int bb = e >> 8;
    const int j  = e & (NN - 1);
    xs[bb * XSTRIDE + j] = x[(size_t)(bchunk0 + bb) * NN + j];
  }
#endif
  {
    const int j = tid;  // 256 threads <-> 256 source nodes
    const float adjv = adjacency[j * NN + i];    // w_ij = adjacency[j,i]
    const float eff  = (adjv > THRESH && j != i) ? adjv : 0.f;
    const int base = (i * NN + j) * PP;
    float* c = &cs[j * 16];
    c[0] = eff * w_pw[base + 0];
    c[1] = eff * w_pw[base + 1];
    c[2] = eff * w_pw[base + 2];
    c[3] = eff * w_pw[base + 3];
    c[4] = bp[base + 0];
    c[5] = bp[base + 1];
    c[6] = bp[base + 2];
    c[7] = bp[base + 3];
    c[8] = eff * w_pw[base + 4];
    c[9] = bp[base + 4];
    red[tid] = eff * b_pw[i * NN + j];
  }
  if (tid < HH) {
    W1s[tid]      = W1[i * HH + tid];
    b1s[tid]      = b1[i * HH + tid];
    W2s[tid]      = W2[(i * 2 + 0) * HH + tid];
    W2s[HH + tid] = W2[(i * 2 + 1) * HH + tid];
    if (tid < 2) b2s[tid] = b2[i * 2 + tid];
  }
#if USE_ASYNC_LDS
  __builtin_amdgcn_s_wait_asynccnt(0);
#endif
  __syncthreads();

  // deterministic bias reduction: bias = sum_j effw*b_pw
  for (int s = 128; s > 0; s >>= 1) {
    if (tid < s) red[tid] += red[tid + s];
    __syncthreads();
  }
  const float bias = red[0];

  // ---------------- Phase 1: contributions via WMMA reduction ----------
  const int lane  = tid & 31;
  const int wid   = tid >> 5;        // 8 waves, one 16-row b-tile each
  const int mrow  = lane & 15;       // A-matrix row (b within tile)
  const int khalf = lane >> 4;       // VGPR half: K = 2*khalf + {0,1}
  const float* xs_row = &xs[(wid * 16 + mrow) * XSTRIDE];

  v2f bones;
  bones.x = bones.y = (mrow == 0) ? 1.0f : 0.0f;  // B: ones in column 0

  v8f acc0 = {}, acc1 = {};          // two streams break the D->C RAW chain
  for (int j0 = 0; j0 < NN; j0 += 8) {
    acc0 = wmma4(pwc_pair(xs_row, cs, j0 + 2 * khalf), bones, acc0);
    acc1 = wmma4(pwc_pair(xs_row, cs, j0 + 4 + 2 * khalf), bones, acc1);
  }
  v8f acc = acc0 + acc1;

  // D column 0 -> contributions; lane L: M = v + 8*khalf, N = L&15
  if (mrow == 0) {
#pragma unroll
    for (int v = 0; v < 8; ++v)
      carr[wid * 16 + 8 * khalf + v] = acc[v] + bias;
  }
  __syncthreads();

  // ---------------- Phase 2: per-node MLP as WMMA GEMM ------------------
  const float cval = carr[wid * 16 + mrow];
  v8f acc2 = {};
#pragma unroll
  for (int s = 0; s < HH / 4; ++s) {
    const int h = s * 4 + 2 * khalf;
    v2f a, b;
    a.x = fmaxf(fmaf(cval, W1s[h],     b1s[h]),     0.f);
    a.y = fmaxf(fmaf(cval, W1s[h + 1], b1s[h + 1]), 0.f);
    const int n = mrow;                 // B column = output index o
    const bool on = (n < 2);
    const int nn = on ? n : 0;
    b.x = on ? W2s[nn * HH + h]     : 0.f;
    b.y = on ? W2s[nn * HH + h + 1] : 0.f;
    acc2 = wmma4(a, b, acc2);
  }

  // D2 col 0 = means, col 1 = log_stds. out = [means (B,N) | log_stds (B,N)]
  if (mrow < 2) {
    const float bb2  = b2s[mrow];
    const size_t obase = (size_t)mrow * (size_t)(BB * NN);
#pragma unroll
    for (int v = 0; v < 8; ++v) {
      const int bglob = bchunk0 + wid * 16 + 8 * khalf + v;
      out[obase + (size_t)bglob * NN + i] = acc2[v] + bb2;
    }
  }
}

extern "C" void kernel_launch(void* const* d_in, const int* in_sizes, int n_in,
                              void* d_out, int out_size, void* d_ws, size_t ws_size,
                              hipStream_t stream) {
  const float* x         = (const float*)d_in[0];
  const float* adjacency = (const float*)d_in[1];
  const float* bp        = (const float*)d_in[2];
  const float* b_pw_dummy= nullptr; (void)b_pw_dummy;
  const float* w_pw      = (const float*)d_in[3];
  const float* b_pw      = (const float*)d_in[4];
  const float* W1        = (const float*)d_in[5];
  const float* b1        = (const float*)d_in[6];
  const float* W2        = (const float*)d_in[7];
  const float* b2        = (const float*)d_in[8];
  float* out = (float*)d_out;

  dim3 grid(NN, BB / BCHUNK);   // (i, batch chunk)
  dim3 block(256);              // 8 wave32
  pwl_mlp_kernel<<<grid, block, 0, stream>>>(
      x, adjacency, bp, w_pw, b_pw, W1, b1, W2, b2, out);
}